// CoSSL_50551765073923
// MI455X (gfx1250) — compile-verified
//
#include <hip/hip_runtime.h>
#include <math.h>

// ---------------------------------------------------------------------------
// CoSSL hard-negative-mining scoring on MI455X (gfx1250, wave32, WMMA).
// Dominant cost: fp32 GEMM 256x65536 (K=2304) streaming 604MB of ref_queue.
// Memory-bound => faithful fp32 V_WMMA_F32_16X16X4_F32. B tile is shared by
// all 4 waves of a block -> stage it ONCE per block in LDS via the gfx1250
// async global->LDS path (ASYNCcnt), double-buffered, fragments via ds_load.
// ---------------------------------------------------------------------------

typedef __attribute__((ext_vector_type(2))) float v2f;
typedef __attribute__((ext_vector_type(8))) float v8f;
typedef __attribute__((ext_vector_type(4))) int   v4i;

#define B_DIM   256
#define D_DIM   128
#define R_DIM   2304
#define Q_DIM   65536
#define TOPK_N  5
#define LDC_OUT (B_DIM + Q_DIM)   // 65792

#define KCHUNK  16                // K rows staged per LDS buffer
#define LDSB    80                // padded LDS row stride (floats): rows 2
                                  // apart differ by 160 mod 64 = +32 banks ->
                                  // the two lane-halves of a fragment read
                                  // disjoint bank sets (conflict-free)

#define AS1 __attribute__((address_space(1)))   // HIP "__device__" (global)
#define AS3 __attribute__((address_space(3)))   // HIP "__shared__" (LDS)

#if __has_builtin(__builtin_amdgcn_global_load_async_to_lds_b128)
#define ASYNC_LDS 1
#if __has_builtin(__builtin_amdgcn_s_wait_asynccnt)
#define ASYNC_WAIT(n) __builtin_amdgcn_s_wait_asynccnt(n)
#else
#define ASYNC_WAIT(n) asm volatile("s_wait_asynccnt %0" ::"n"(n) : "memory")
#endif
#else
#define ASYNC_LDS 0
#define ASYNC_WAIT(n) ((void)0)   // sync fallback: __syncthreads covers DScnt
#endif

// ---------------------------------------------------------------------------
// 1) L2-normalize q rows -> qn [B,D]; normalize k rows and store transposed
//    knT [D,B] so score_batch can reuse the generic row-major GEMM.
// ---------------------------------------------------------------------------
__global__ void normalize_qk(const float* __restrict__ q,
                             const float* __restrict__ k,
                             float* __restrict__ qn,
                             float* __restrict__ knT) {
  const int row = blockIdx.x;      // 0..255
  const int d   = threadIdx.x;     // 0..127
  __shared__ float red[D_DIM];

  const float qv = q[row * D_DIM + d];
  const float kv = k[row * D_DIM + d];

  red[d] = qv * qv;
  __syncthreads();
  for (int s = D_DIM / 2; s > 0; s >>= 1) {
    if (d < s) red[d] += red[d + s];
    __syncthreads();
  }
  const float qnorm = sqrtf(red[0]);
  __syncthreads();

  red[d] = kv * kv;
  __syncthreads();
  for (int s = D_DIM / 2; s > 0; s >>= 1) {
    if (d < s) red[d] += red[d + s];
    __syncthreads();
  }
  const float knorm = sqrtf(red[0]);

  qn[row * D_DIM + d]  = qv / qnorm;
  knT[d * B_DIM + row] = kv / knorm;   // transposed for GEMM B-operand
}

// ---------------------------------------------------------------------------
// 2) fp32 WMMA GEMM with LDS-staged B: C[r, colOff+c] = sum_k A[r,k]*Bm[k,c]
//    Block = 128 threads = 4 waves; block tile 64 rows x 64 cols; each wave
//    owns a 16x64 slab (4 v8f accumulators). B chunk (16 x 64) staged in LDS
//    once per block (async if available), double-buffered.
//    WMMA VGPR layouts per CDNA5 ISA §7.12.2:
//      A 16x4 : lanes 0-15 -> M=lane, VGPR{0,1}=K{0,1}; lanes 16-31 -> K{2,3}
//      B 4x16 : VGPR0 = rows K{0(lo half),2(hi half)}, VGPR1 = K{1,3}
//      C/D    : VGPR r -> M = r (lanes 0-15) / r+8 (lanes 16-31), N = lane%16
// ---------------------------------------------------------------------------
__global__ void wmma_gemm_f32(const float* __restrict__ A,
                              const float* __restrict__ Bm,
                              float* __restrict__ C,
                              int K, int ldb, int ldc, int colOff) {
  const int tid  = threadIdx.x;
  const int lane = tid & 31;
  const int wave = tid >> 5;                  // 0..3
  const int half = lane >> 4;                 // 0 or 1
  const int ml   = lane & 15;

  const int r0 = blockIdx.y * 64 + wave * 16; // row slab base for this wave
  const int n0 = blockIdx.x * 64;             // col tile base for this block

  __shared__ float lds[2][KCHUNK * LDSB];

  // staging assignment: thread t covers B rows (t>>4) and 8+(t>>4),
  // columns 4*(t&15)..+3 of the 16x64 chunk (one b128 each).
  const int srow = tid >> 4;                  // 0..7
  const int scol = (tid & 15) * 4;            // 0,4,...,60

  v8f acc0 = {}, acc1 = {}, acc2 = {}, acc3 = {};
  const float* Arow = A + (size_t)(r0 + ml) * K;
  const int nchunk = K / KCHUNK;

  // ---- stage one 16x64 chunk of B into lds[buf] -------------------------
  auto stage = [&](int chunk, int buf) {
    const int kb = chunk * KCHUNK;
    const float* g0 = Bm + (size_t)(kb + srow) * ldb + n0 + scol;
    const float* g1 = Bm + (size_t)(kb + 8 + srow) * ldb + n0 + scol;
    float* l0 = &lds[buf][srow * LDSB + scol];
    float* l1 = &lds[buf][(8 + srow) * LDSB + scol];
#if ASYNC_LDS
    __builtin_amdgcn_global_load_async_to_lds_b128((AS1 v4i*)g0, (AS3 v4i*)l0, 0, 0);
    __builtin_amdgcn_global_load_async_to_lds_b128((AS1 v4i*)g1, (AS3 v4i*)l1, 0, 0);
#else
    *(float4*)l0 = *(const float4*)g0;
    *(float4*)l1 = *(const float4*)g1;
#endif
  };

  stage(0, 0);                                 // prologue: chunk 0 -> buf 0

  for (int c = 0; c < nchunk; ++c) {
    const int buf = c & 1;
    if (c + 1 < nchunk) {
      stage(c + 1, buf ^ 1);                   // overlap next fetch
      ASYNC_WAIT(2);                           // chunk c complete (in-order)
    } else {
      ASYNC_WAIT(0);
    }
    __syncthreads();                           // chunk c visible to all waves

    const float* Lb = &lds[buf][0];
    const int k0 = c * KCHUNK;
#pragma unroll
    for (int kk = 0; kk < KCHUNK; kk += 4) {
      v2f a = *(const v2f*)(Arow + k0 + kk + 2 * half);
      const float* bb = Lb + (kk + 2 * half) * LDSB + ml;
      v2f b0, b1, b2, b3;
      b0.x = bb[0];   b0.y = bb[LDSB];
      b1.x = bb[16];  b1.y = bb[LDSB + 16];
      b2.x = bb[32];  b2.y = bb[LDSB + 32];
      b3.x = bb[48];  b3.y = bb[LDSB + 48];
      acc0 = __builtin_amdgcn_wmma_f32_16x16x4_f32(false, a, false, b0,
                                                   (short)0, acc0, false, false);
      acc1 = __builtin_amdgcn_wmma_f32_16x16x4_f32(false, a, false, b1,
                                                   (short)0, acc1, false, false);
      acc2 = __builtin_amdgcn_wmma_f32_16x16x4_f32(false, a, false, b2,
                                                   (short)0, acc2, false, false);
      acc3 = __builtin_amdgcn_wmma_f32_16x16x4_f32(false, a, false, b3,
                                                   (short)0, acc3, false, false);
    }
    __syncthreads();                           // all waves done with lds[buf]
  }

  // Store: VGPR r -> row r0 + r + 8*half, col n0 + ml (+16*tile)
  float* Cbase = C + (size_t)(r0 + 8 * half) * ldc + colOff + n0 + ml;
#pragma unroll
  for (int r = 0; r < 8; ++r) {
    float* crow = Cbase + (size_t)r * ldc;
    crow[0]  = acc0[r];
    crow[16] = acc1[r];
    crow[32] = acc2[r];
    crow[48] = acc3[r];
  }
}

// ---------------------------------------------------------------------------
// 3) Per-row masked top-5 of score_ref (same-audio entries -> -inf).
//    256 threads/row keep private sorted top-5, then LDS tree-merge.
// ---------------------------------------------------------------------------
__global__ void topk_kernel(const float* __restrict__ score_ref,
                            const int* __restrict__ indices,
                            const int* __restrict__ index_queue,
                            int* __restrict__ topk_idx) {
  const int b = blockIdx.x;
  const int t = threadIdx.x;                  // 0..255
  const int my = indices[b];
  const float NEG_INF = -__builtin_inff();

  float vals[TOPK_N];
  int   idxs[TOPK_N];
#pragma unroll
  for (int i = 0; i < TOPK_N; ++i) { vals[i] = NEG_INF; idxs[i] = -1; }

  const float* row = score_ref + (size_t)b * Q_DIM;
  for (int qx = t; qx < Q_DIM; qx += 256) {
    float v = (index_queue[qx] == my) ? NEG_INF : row[qx];
    if (v > vals[TOPK_N - 1]) {
      int p = TOPK_N - 1;
      while (p > 0 && v > vals[p - 1]) {
        vals[p] = vals[p - 1]; idxs[p] = idxs[p - 1]; --p;
      }
      vals[p] = v; idxs[p] = qx;
    }
  }

  __shared__ float sv[256 * TOPK_N];
  __shared__ int   si[256 * TOPK_N];
#pragma unroll
  for (int i = 0; i < TOPK_N; ++i) { sv[t * TOPK_N + i] = vals[i]; si[t * TOPK_N + i] = idxs[i]; }
  __syncthreads();

  for (int s = 128; s > 0; s >>= 1) {
    if (t < s) {
      float* av = &sv[t * TOPK_N];        int* ai = &si[t * TOPK_N];
      float* bv = &sv[(t + s) * TOPK_N];  int* bi = &si[(t + s) * TOPK_N];
      float mv[TOPK_N]; int mi[TOPK_N];
      int ia = 0, ib = 0;
#pragma unroll
      for (int o = 0; o < TOPK_N; ++o) {
        if (av[ia] >= bv[ib]) { mv[o] = av[ia]; mi[o] = ai[ia]; ++ia; }
        else                  { mv[o] = bv[ib]; mi[o] = bi[ib]; ++ib; }
      }
#pragma unroll
      for (int o = 0; o < TOPK_N; ++o) { av[o] = mv[o]; ai[o] = mi[o]; }
    }
    __syncthreads();
  }
  if (t < TOPK_N) topk_idx[b * TOPK_N + t] = si[t];
}

// ---------------------------------------------------------------------------
// 4) Compose final score & mask.
//    score cols [0,256)  : already score_batch (written by GEMM).
//    score cols [256,..) : score_queue_raw (in d_out) * score_ref * wmask.
//    mask written as float into second half of d_out.
// ---------------------------------------------------------------------------
__global__ void compose_kernel(const float* __restrict__ score_ref,
                               const int* __restrict__ indices,
                               const int* __restrict__ index_queue,
                               const int* __restrict__ topk_idx,
                               float* __restrict__ out) {
  const int b = blockIdx.y;
  const int j = blockIdx.x * blockDim.x + threadIdx.x;   // 0..65791
  if (j >= LDC_OUT) return;

  float* score = out;
  float* mask  = out + (size_t)B_DIM * LDC_OUT;
  const size_t off = (size_t)b * LDC_OUT + j;
  const int my = indices[b];

  if (j < B_DIM) {
    mask[off] = (my == indices[j]) ? 1.0f : 0.0f;
  } else {
    const int qx = j - B_DIM;
    const float sq = score[off];                       // raw qn @ moco_queue
    const float sr = score_ref[(size_t)b * Q_DIM + qx];
    bool isTop = false;
#pragma unroll
    for (int i = 0; i < TOPK_N; ++i) isTop |= (topk_idx[b * TOPK_N + i] == qx);
    const float wm = isTop ? 1.0f : -1.0f;
    score[off] = sq * sr * wm;
    const bool eq = (index_queue[qx] == my);
    mask[off] = (isTop || eq) ? 1.0f : 0.0f;
  }
}

// ---------------------------------------------------------------------------
extern "C" void kernel_launch(void* const* d_in, const int* in_sizes, int n_in,
                              void* d_out, int out_size, void* d_ws, size_t ws_size,
                              hipStream_t stream) {
  const float* q           = (const float*)d_in[0];
  const float* k           = (const float*)d_in[1];
  const float* ref_feats   = (const float*)d_in[2];
  const float* moco_queue  = (const float*)d_in[3];
  const float* ref_queue   = (const float*)d_in[4];
  const int*   indices     = (const int*)d_in[5];
  const int*   index_queue = (const int*)d_in[6];
  float* out = (float*)d_out;

  // Workspace layout (floats): qn[32768] | knT[32768] | score_ref[16.7M] | topk
  float* qn        = (float*)d_ws;
  float* knT       = qn + B_DIM * D_DIM;
  float* score_ref = knT + D_DIM * B_DIM;
  int*   topk_idx  = (int*)(score_ref + (size_t)B_DIM * Q_DIM);

  normalize_qk<<<B_DIM, D_DIM, 0, stream>>>(q, k, qn, knT);

  const dim3 blk(128);
  // score_ref = ref_feats @ ref_queue   [256 x 65536], K = 2304 (dominant)
  wmma_gemm_f32<<<dim3(Q_DIM / 64, B_DIM / 64), blk, 0, stream>>>(
      ref_feats, ref_queue, score_ref, R_DIM, Q_DIM, Q_DIM, 0);
  // score_queue_raw = qn @ moco_queue -> d_out cols [256, 65792)
  wmma_gemm_f32<<<dim3(Q_DIM / 64, B_DIM / 64), blk, 0, stream>>>(
      qn, moco_queue, out, D_DIM, Q_DIM, LDC_OUT, B_DIM);
  // score_batch = qn @ knT -> d_out cols [0, 256)
  wmma_gemm_f32<<<dim3(B_DIM / 64, B_DIM / 64), blk, 0, stream>>>(
      qn, knT, out, D_DIM, B_DIM, LDC_OUT, 0);

  topk_kernel<<<B_DIM, 256, 0, stream>>>(score_ref, indices, index_queue, topk_idx);

  compose_kernel<<<dim3(LDC_OUT / 256, B_DIM), 256, 0, stream>>>(
      score_ref, indices, index_queue, topk_idx, out);
}